// UGFMEncoder_40956808134666
// MI455X (gfx1250) — compile-verified
//
#include <hip/hip_runtime.h>
#include <hip/hip_bf16.h>
#include <math.h>

#define D_MODEL 128
#define NHEAD 8
#define N_NODES 20000
#define N_EDGES 200000
#define M_NODES 4000
#define M_EDGES 20000
#define N_TARGETS 512
#define N_KEYS 6
// param-key rows: TW, NB, NQ, EB, EK, EV
#define ROW_TW 0
#define ROW_NB 1
#define ROW_NQ 2
#define ROW_EB 3
#define ROW_EK 4
#define ROW_EV 5

typedef float v2f __attribute__((ext_vector_type(2)));
typedef float v8f __attribute__((ext_vector_type(8)));

__device__ __forceinline__ v8f wmma_f32x4(v2f a, v2f b, v8f c) {
  // D = A(16x4 f32) * B(4x16 f32) + C(16x16 f32), exact fp32
  return __builtin_amdgcn_wmma_f32_16x16x4_f32(false, a, false, b, (short)0, c, false, false);
}

__device__ __forceinline__ unsigned f2ord(float f) {
  unsigned u = __float_as_uint(f);
  return (u & 0x80000000u) ? ~u : (u | 0x80000000u);
}
__device__ __forceinline__ float ord2f(unsigned o) {
  return (o & 0x80000000u) ? __uint_as_float(o & 0x7fffffffu) : __uint_as_float(~o);
}

// ---------------------------------------------------------------------------
// meta_learner: softmax(feat @ u^T / sqrt(D)) @ u, then @W2 -> p2, @W1 -> p1
// One wave per 16-row tile; all matmuls via v_wmma_f32_16x16x4_f32.
// ---------------------------------------------------------------------------
__global__ __launch_bounds__(128)
void meta_learner_kernel(const float* __restrict__ feat, int nrows,
                         const float* __restrict__ u,
                         const float* __restrict__ W2,
                         const float* __restrict__ W1,
                         float* __restrict__ p2out,
                         float* __restrict__ p1out)
{
  __shared__ float slds[4][16 * 16];    // softmax tile, per wave
  __shared__ float zlds[4][16 * 128];   // z tile, per wave

  const int lane = threadIdx.x & 31;
  const int w = threadIdx.x >> 5;
  const int tile = blockIdx.x * 4 + w;
  const int row0 = tile * 16;
  if (row0 >= nrows) return;            // uniform per wave

  const int hp  = lane >> 4;            // half: 0 or 1
  const int l16 = lane & 15;
  float* sw = slds[w];
  float* zw = zlds[w];

  int arow = row0 + l16;
  if (arow >= nrows) arow = nrows - 1;  // clamp reads; writes guarded
  const float* frow = feat + (size_t)arow * D_MODEL;
  const float* urow = u + (size_t)l16 * D_MODEL;   // column l16 of u^T

  // ---- logits[16x16] = feat_tile @ u^T ----
  v8f acc = {0.f,0.f,0.f,0.f,0.f,0.f,0.f,0.f};
  for (int k0 = 0; k0 < D_MODEL; k0 += 4) {
    v2f a = *(const v2f*)(frow + k0 + 2 * hp);
    v2f b = *(const v2f*)(urow + k0 + 2 * hp);
    acc = wmma_f32x4(a, b, acc);
  }

  // ---- softmax over latent dim (the 16 lanes of each half) ----
  const float scale = 0.08838834764831845f;  // 1/sqrt(128)
  #pragma unroll
  for (int v = 0; v < 8; ++v) {
    float x = acc[v] * scale;
    float mx = x;
    #pragma unroll
    for (int sh = 1; sh < 16; sh <<= 1) mx = fmaxf(mx, __shfl_xor(mx, sh, 32));
    float e = __expf(x - mx);
    float sm = e;
    #pragma unroll
    for (int sh = 1; sh < 16; sh <<= 1) sm += __shfl_xor(sm, sh, 32);
    sw[(v + 8 * hp) * 16 + l16] = e / sm;    // row-major s[m][l]
  }
  asm volatile("s_wait_dscnt 0" ::: "memory");

  // ---- z[16x128] = s @ u ----
  for (int dt = 0; dt < 8; ++dt) {
    const int dcol = dt * 16 + l16;
    v8f az = {0.f,0.f,0.f,0.f,0.f,0.f,0.f,0.f};
    #pragma unroll
    for (int k0 = 0; k0 < 16; k0 += 4) {
      v2f a = *(const v2f*)(&sw[l16 * 16 + k0 + 2 * hp]);
      v2f b;
      b.x = u[(size_t)(k0 + 2 * hp) * D_MODEL + dcol];
      b.y = u[(size_t)(k0 + 2 * hp + 1) * D_MODEL + dcol];
      az = wmma_f32x4(a, b, az);
    }
    #pragma unroll
    for (int v = 0; v < 8; ++v)
      zw[(v + 8 * hp) * D_MODEL + dcol] = az[v];
  }
  asm volatile("s_wait_dscnt 0" ::: "memory");

  // ---- p2 = z @ W2 ; p1 = z @ W1 ----
  for (int which = 0; which < 2; ++which) {
    const float* W = which ? W1 : W2;
    float* out = which ? p1out : p2out;
    for (int dt = 0; dt < 8; ++dt) {
      const int dcol = dt * 16 + l16;
      v8f c = {0.f,0.f,0.f,0.f,0.f,0.f,0.f,0.f};
      for (int k0 = 0; k0 < D_MODEL; k0 += 4) {
        v2f a = *(const v2f*)(&zw[l16 * D_MODEL + k0 + 2 * hp]);
        v2f b;
        b.x = W[(size_t)(k0 + 2 * hp) * D_MODEL + dcol];
        b.y = W[(size_t)(k0 + 2 * hp + 1) * D_MODEL + dcol];
        c = wmma_f32x4(a, b, c);
      }
      #pragma unroll
      for (int v = 0; v < 8; ++v) {
        int orow = row0 + v + 8 * hp;
        if (orow < nrows) out[(size_t)orow * D_MODEL + dcol] = c[v];
      }
    }
  }
}

// ---------------------------------------------------------------------------
// Misc elementwise kernels
// ---------------------------------------------------------------------------
__global__ void key_feats_kernel(const float* __restrict__ emb, const int* __restrict__ kv,
                                 const int* __restrict__ kt, float* __restrict__ out)
{
  int k = blockIdx.x, d = threadIdx.x;
  float v = emb[(size_t)kv[k] * D_MODEL + d];
  for (int t = 0; t < 3; ++t) v += emb[(size_t)kt[k * 3 + t] * D_MODEL + d];
  out[k * D_MODEL + d] = v;
}

__global__ void gather_rows_kernel(const float* __restrict__ emb, const int* __restrict__ ids,
                                   float* __restrict__ out, int n)
{
  long g = (long)blockIdx.x * blockDim.x + threadIdx.x;
  if (g >= (long)n * D_MODEL) return;
  out[g] = emb[(size_t)ids[g >> 7] * D_MODEL + (g & 127)];
}

__global__ void copy_kernel(float* __restrict__ d, const float* __restrict__ s, long n) {
  long g = (long)blockIdx.x * blockDim.x + threadIdx.x;
  if (g < n) d[g] = s[g];
}
__global__ void add_kernel(float* __restrict__ a, const float* __restrict__ b, long n) {
  long g = (long)blockIdx.x * blockDim.x + threadIdx.x;
  if (g < n) a[g] += b[g];
}

// ---------------------------------------------------------------------------
// hetero_conv phases
// ---------------------------------------------------------------------------
__global__ void conv_init_kernel(unsigned* __restrict__ mvals, float* __restrict__ denom,
                                 float* __restrict__ agg, int nH, long nD)
{
  long i = (long)blockIdx.x * blockDim.x + threadIdx.x;
  if (i < nH) { mvals[i] = 0x007FFFFFu; /* f2ord(-inf) */ denom[i] = 0.f; }
  if (i < nD) agg[i] = 0.f;
}

__global__ void conv_logits_kernel(const float* __restrict__ h, const int* __restrict__ src,
                                   const int* __restrict__ dst, int E,
                                   const float* __restrict__ bp_nq, const float* __restrict__ bp_ek,
                                   const float* p2n, const int* mnid,
                                   const float* p2e, const int* meid,
                                   float* __restrict__ logits, unsigned* __restrict__ mvals)
{
  long g = (long)blockIdx.x * blockDim.x + threadIdx.x;
  int e = (int)(g >> 7);
  if (e >= E) return;
  int t = (int)(g & 127);
  int head = t >> 4;
  int s = src[e], d = dst[e];
  float nq = bp_nq[t]; if (p2n) nq += p2n[(size_t)mnid[d] * D_MODEL + t];
  float ek = bp_ek[t]; if (p2e) ek += p2e[(size_t)meid[e] * D_MODEL + t];
  float prod = (h[(size_t)d * D_MODEL + t] * nq) * (h[(size_t)s * D_MODEL + t] * ek);
  #pragma unroll
  for (int m = 1; m < 16; m <<= 1) prod += __shfl_xor(prod, m, 32);
  if ((t & 15) == 0) {
    float lg = prod * 0.25f;  // 1/sqrt(16)
    logits[(size_t)e * NHEAD + head] = lg;
    atomicMax(&mvals[(size_t)d * NHEAD + head], f2ord(lg));
  }
}

__global__ void conv_finmax_kernel(unsigned* mvals, float* m, int n)
{
  int i = blockIdx.x * blockDim.x + threadIdx.x;
  if (i >= n) return;
  float f = ord2f(mvals[i]);
  if (!isfinite(f)) f = 0.f;
  m[i] = f;   // aliases mvals (same buffer reinterpreted)
}

__global__ void conv_exp_kernel(const int* __restrict__ dst, int E,
                                const float* __restrict__ m, float* __restrict__ logits,
                                float* __restrict__ denom)
{
  long g = (long)blockIdx.x * blockDim.x + threadIdx.x;
  if (g >= (long)E * NHEAD) return;
  int e = (int)(g >> 3), hh = (int)(g & 7);
  float ev = __expf(logits[g] - m[(size_t)dst[e] * NHEAD + hh]);
  logits[g] = ev;
  atomicAdd(&denom[(size_t)dst[e] * NHEAD + hh], ev);
}

__global__ void conv_msg_kernel(const float* __restrict__ h, const int* __restrict__ src,
                                const int* __restrict__ dst, int E,
                                const float* __restrict__ bp_ev, const float* __restrict__ bp_eb,
                                const float* p2e, const float* p1e, const int* meid,
                                const float* __restrict__ eexp, const float* __restrict__ denom,
                                float* __restrict__ agg, float* edge_feat)
{
  long g = (long)blockIdx.x * blockDim.x + threadIdx.x;
  int e = (int)(g >> 7);
  if (e >= E) return;
  int t = (int)(g & 127);
  int head = t >> 4;
  int s = src[e], d = dst[e];
  float ev = bp_ev[t]; if (p2e) ev += p2e[(size_t)meid[e] * D_MODEL + t];
  float attn = eexp[(size_t)e * NHEAD + head] / (denom[(size_t)d * NHEAD + head] + 1e-9f);
  float msg = attn * (h[(size_t)s * D_MODEL + t] * ev);
  atomicAdd(&agg[(size_t)d * D_MODEL + t], msg);
  if (edge_feat) {
    float eb = bp_eb[t]; if (p1e) eb += p1e[(size_t)meid[e] * D_MODEL + t];
    edge_feat[(size_t)e * D_MODEL + t] = fmaxf(msg + eb, 0.f);
  }
}

__global__ void conv_out_kernel(const float* __restrict__ agg, const float* __restrict__ bp_nb,
                                const float* p1n, const int* mnid,
                                float* __restrict__ hout, int N)
{
  long g = (long)blockIdx.x * blockDim.x + threadIdx.x;
  if (g >= (long)N * D_MODEL) return;
  int n = (int)(g >> 7), t = (int)(g & 127);
  float nb = bp_nb[t]; if (p1n) nb += p1n[(size_t)mnid[n] * D_MODEL + t];
  hout[g] = fmaxf(agg[g] + nb, 0.f);
}

__global__ void readout_kernel(const float* __restrict__ h, const int* __restrict__ target_idx,
                               const float* __restrict__ bp_tw, const float* __restrict__ p1n,
                               const int* __restrict__ mnid, float* __restrict__ out, int accumulate)
{
  int t = blockIdx.x, d = threadIdx.x;
  int n = target_idx[t];
  float tw = bp_tw[d] + p1n[(size_t)mnid[n] * D_MODEL + d];
  float val = h[(size_t)n * D_MODEL + d] * tw;
  #pragma unroll
  for (int m = 1; m < 32; m <<= 1) val += __shfl_xor(val, m, 32);
  __shared__ float red[4];
  if ((threadIdx.x & 31) == 0) red[threadIdx.x >> 5] = val;
  __syncthreads();
  if (threadIdx.x == 0) {
    float s = red[0] + red[1] + red[2] + red[3];
    out[t] = accumulate ? (out[t] + 0.5f * s) : (0.5f * s);
  }
}

// ---------------------------------------------------------------------------
// Host orchestration
// ---------------------------------------------------------------------------
static inline int divup(long a, long b) { return (int)((a + b - 1) / b); }

static void run_conv(const float* h, const int* src, const int* dst, int E, int N,
                     const float* bp_nq, const float* bp_nb, const float* bp_ek,
                     const float* bp_ev, const float* bp_eb,
                     const float* p2n, const float* p1n, const int* mnid,
                     const float* p2e, const float* p1e, const int* meid,
                     float* logits, unsigned* mvals, float* denom, float* agg,
                     float* edge_feat, float* hout, hipStream_t stream)
{
  long nD = (long)N * D_MODEL;
  long eT = (long)E * D_MODEL;
  conv_init_kernel<<<divup(nD, 256), 256, 0, stream>>>(mvals, denom, agg, N * NHEAD, nD);
  conv_logits_kernel<<<divup(eT, 256), 256, 0, stream>>>(h, src, dst, E, bp_nq, bp_ek,
                                                         p2n, mnid, p2e, meid, logits, mvals);
  conv_finmax_kernel<<<divup((long)N * NHEAD, 256), 256, 0, stream>>>(mvals, (float*)mvals, N * NHEAD);
  conv_exp_kernel<<<divup((long)E * NHEAD, 256), 256, 0, stream>>>(dst, E, (const float*)mvals, logits, denom);
  conv_msg_kernel<<<divup(eT, 256), 256, 0, stream>>>(h, src, dst, E, bp_ev, bp_eb,
                                                      p2e, p1e, meid, logits, denom, agg, edge_feat);
  conv_out_kernel<<<divup(nD, 256), 256, 0, stream>>>(agg, bp_nb, p1n, mnid, hout, N);
}

extern "C" void kernel_launch(void* const* d_in, const int* in_sizes, int n_in,
                              void* d_out, int out_size, void* d_ws, size_t ws_size,
                              hipStream_t stream)
{
  (void)in_sizes; (void)n_in; (void)out_size; (void)ws_size;
  const int* node_value_ids = (const int*)d_in[0];
  const int* meta_node_id   = (const int*)d_in[1];
  const int* src            = (const int*)d_in[2];
  const int* dst            = (const int*)d_in[3];
  const int* meta_edge_id   = (const int*)d_in[4];
  const int* meta_value_ids = (const int*)d_in[5];
  const int* meta_src       = (const int*)d_in[6];
  const int* meta_dst       = (const int*)d_in[7];
  const int* target_idx     = (const int*)d_in[8];
  const int* key_value_ids  = (const int*)d_in[9];
  const int* key_token_ids  = (const int*)d_in[10];
  const float* emb = (const float*)d_in[11];
  const float* u   = (const float*)d_in[12];
  const float* W2  = (const float*)d_in[13];
  const float* W1  = (const float*)d_in[14];
  float* out = (float*)d_out;

  float* p = (float*)d_ws;
  float* key_feats = p;        p += N_KEYS * D_MODEL;
  float* bp2 = p;              p += N_KEYS * D_MODEL;
  float* bp1 = p;              p += N_KEYS * D_MODEL;
  float* meta_feat = p;        p += (long)M_NODES * D_MODEL;
  float* node_feat = p;        p += (long)N_NODES * D_MODEL;
  float* pre_meta = p;         p += (long)M_NODES * D_MODEL;
  float* pre_node = p;         p += (long)N_NODES * D_MODEL;
  float* meta_edge_feat = p;   p += (long)M_EDGES * D_MODEL;
  float* p2n = p;              p += (long)M_NODES * D_MODEL;
  float* p1n = p;              p += (long)M_NODES * D_MODEL;
  float* p2e = p;              p += (long)M_EDGES * D_MODEL;
  float* p1e = p;              p += (long)M_EDGES * D_MODEL;
  float* logits = p;           p += (long)N_EDGES * NHEAD;
  unsigned* mvals = (unsigned*)p; p += (long)N_NODES * NHEAD;
  float* denom = p;            p += (long)N_NODES * NHEAD;
  float* agg = p;              p += (long)N_NODES * D_MODEL;

  // Base params from param-key graph features
  key_feats_kernel<<<N_KEYS, 128, 0, stream>>>(emb, key_value_ids, key_token_ids, key_feats);
  meta_learner_kernel<<<1, 128, 0, stream>>>(key_feats, N_KEYS, u, W2, W1, bp2, bp1);

  // Embedding gathers
  gather_rows_kernel<<<divup((long)M_NODES * D_MODEL, 256), 256, 0, stream>>>(emb, meta_value_ids, meta_feat, M_NODES);
  gather_rows_kernel<<<divup((long)N_NODES * D_MODEL, 256), 256, 0, stream>>>(emb, node_value_ids, node_feat, N_NODES);

  const float* bp_nq = bp2 + ROW_NQ * D_MODEL;
  const float* bp_ek = bp2 + ROW_EK * D_MODEL;
  const float* bp_ev = bp2 + ROW_EV * D_MODEL;
  const float* bp_nb = bp1 + ROW_NB * D_MODEL;
  const float* bp_eb = bp1 + ROW_EB * D_MODEL;
  const float* bp_tw = bp1 + ROW_TW * D_MODEL;

  for (int blk = 0; blk < 2; ++blk) {
    copy_kernel<<<divup((long)M_NODES * D_MODEL, 256), 256, 0, stream>>>(pre_meta, meta_feat, (long)M_NODES * D_MODEL);
    copy_kernel<<<divup((long)N_NODES * D_MODEL, 256), 256, 0, stream>>>(pre_node, node_feat, (long)N_NODES * D_MODEL);

    // Meta-graph conv (broadcast base params), in-place on meta_feat
    run_conv(meta_feat, meta_src, meta_dst, M_EDGES, M_NODES,
             bp_nq, bp_nb, bp_ek, bp_ev, bp_eb,
             nullptr, nullptr, nullptr, nullptr, nullptr, nullptr,
             logits, mvals, denom, agg, meta_edge_feat, meta_feat, stream);

    // Meta-learned per-meta params (WMMA GEMMs)
    {
      int tiles = (M_NODES + 15) / 16;
      meta_learner_kernel<<<divup(tiles, 4), 128, 0, stream>>>(meta_feat, M_NODES, u, W2, W1, p2n, p1n);
      tiles = (M_EDGES + 15) / 16;
      meta_learner_kernel<<<divup(tiles, 4), 128, 0, stream>>>(meta_edge_feat, M_EDGES, u, W2, W1, p2e, p1e);
    }

    // Node-graph conv (base + gathered meta params), in-place on node_feat
    run_conv(node_feat, src, dst, N_EDGES, N_NODES,
             bp_nq, bp_nb, bp_ek, bp_ev, bp_eb,
             p2n, p1n, meta_node_id, p2e, p1e, meta_edge_id,
             logits, mvals, denom, agg, nullptr, node_feat, stream);

    // Target readout (accumulates the mean over the two blocks)
    readout_kernel<<<N_TARGETS, 128, 0, stream>>>(node_feat, target_idx, bp_tw, p1n, meta_node_id, out, blk);

    // Block residuals
    add_kernel<<<divup((long)M_NODES * D_MODEL, 256), 256, 0, stream>>>(meta_feat, pre_meta, (long)M_NODES * D_MODEL);
    add_kernel<<<divup((long)N_NODES * D_MODEL, 256), 256, 0, stream>>>(node_feat, pre_node, (long)N_NODES * D_MODEL);
  }
}